// edge_mask_7000796692946
// MI455X (gfx1250) — compile-verified
//
#include <hip/hip_runtime.h>
#include <math.h>

// ---------------------------------------------------------------------------
// GCN link-predictor pipeline for MI455X (gfx1250, wave32).
// All GEMMs run on V_WMMA_F32_16X16X4_F32 (exact fp32 matrix math).
// Each wave owns 5 m-tiles (5 independent accumulator chains) so B-operand
// (weight) fragments are loaded once per K-step and reused 5x, and the WMMA
// issue stream has no D->C RAW chain between consecutive matrix ops.
// Scatter phases use global_atomic_add_f32; BN stats via LDS ds_add_f32.
// ---------------------------------------------------------------------------

typedef __attribute__((ext_vector_type(2))) float v2f;
typedef __attribute__((ext_vector_type(8))) float v8f;

#define N_NODES 50000
#define N_EDGES 800000
#define IN_D    256
#define HID     128
#define OUTF    64
#define BN_EPS  1e-5f
#define MPW     5            // m-tiles per wave

__device__ __forceinline__ v8f wmma_f32(v2f a, v2f b, v8f c) {
  // D = A(16x4,f32) * B(4x16,f32) + C(16x16,f32)
  return __builtin_amdgcn_wmma_f32_16x16x4_f32(
      /*neg_a=*/false, a, /*neg_b=*/false, b,
      /*c_mod=*/(short)0, c, /*reuse_a=*/false, /*reuse_b=*/false);
}

// ---------------------------------------------------------------------------
// Degree / normalization
// ---------------------------------------------------------------------------
__global__ void deg_init_kernel(float* __restrict__ deg) {
  int i = blockIdx.x * blockDim.x + threadIdx.x;
  if (i < N_NODES) deg[i] = 1.0f;  // self loop
}

__global__ void deg_count_kernel(const int* __restrict__ dst, float* __restrict__ deg) {
  int e = blockIdx.x * blockDim.x + threadIdx.x;
  if (e < N_EDGES) atomicAdd(&deg[dst[e]], 1.0f);
}

__global__ void dinv_kernel(const float* __restrict__ deg, float* __restrict__ dinv) {
  int i = blockIdx.x * blockDim.x + threadIdx.x;
  if (i < N_NODES) dinv[i] = rsqrtf(deg[i]);  // deg >= 1 always
}

// ---------------------------------------------------------------------------
// Node GEMM: C[M,NC] = (reluIn? relu(A) : A)[M,K] @ W[K,NC] (+ bias)
// One wave computes MPW vertically-stacked 16x16 C tiles.
// A-frag (16x4 f32): lanes 0-15 rows M0..15 K={0,1}; lanes 16-31 K={2,3}.
// B-frag (4x16 f32): lanes 0-15 cols N0..15 K={0,1}; lanes 16-31 K={2,3}.
// C/D: VGPR r -> row r (lanes 0-15) / row 8+r (lanes 16-31).
// ---------------------------------------------------------------------------
template <int K, int NC, bool RELU_IN, bool BIAS>
__global__ __launch_bounds__(256) void node_gemm_kernel(
    const float* __restrict__ A, const float* __restrict__ W,
    const float* __restrict__ bias, float* __restrict__ C, int M) {
  const int lane    = threadIdx.x & 31;
  const int waveIdx = blockIdx.x * 8 + (threadIdx.x >> 5);
  const int mBase   = waveIdx * MPW;                 // first m-tile of this wave
  if (mBase * 16 >= M) return;                       // wave-uniform: EXEC stays all-1s
  const int col = blockIdx.y * 16 + (lane & 15);
  const int kh  = (lane >> 4) * 2;                   // 0 or 2

  const float* __restrict__ ap[MPW];
#pragma unroll
  for (int i = 0; i < MPW; ++i)
    ap[i] = A + (size_t)((mBase + i) * 16 + (lane & 15)) * K;

  v8f acc[MPW];
#pragma unroll
  for (int i = 0; i < MPW; ++i) acc[i] = (v8f){};

#pragma unroll
  for (int k = 0; k < K; k += 4) {
    v2f b = {W[(size_t)(k + kh) * NC + col], W[(size_t)(k + kh + 1) * NC + col]};
#pragma unroll
    for (int i = 0; i < MPW; ++i) {
      float a0 = ap[i][k + kh];
      float a1 = ap[i][k + kh + 1];
      if (RELU_IN) { a0 = fmaxf(a0, 0.0f); a1 = fmaxf(a1, 0.0f); }
      v2f a = {a0, a1};
      acc[i] = wmma_f32(a, b, acc[i]);
    }
  }
  const float bb = BIAS ? bias[col] : 0.0f;
#pragma unroll
  for (int i = 0; i < MPW; ++i) {
    const int rbase = (mBase + i) * 16 + (lane >> 4) * 8;
#pragma unroll
    for (int r = 0; r < 8; ++r)
      C[(size_t)(rbase + r) * NC + col] = acc[i][r] + bb;
  }
}

// ---------------------------------------------------------------------------
// agg[n,f] = t[n,f] * dinv[n]^2 + b[f]   (self-loop term + bias, also zero-init)
// ---------------------------------------------------------------------------
template <int F>
__global__ void self_init_kernel(const float* __restrict__ t,
                                 const float* __restrict__ dinv,
                                 const float* __restrict__ b,
                                 float* __restrict__ agg) {
  int idx = blockIdx.x * blockDim.x + threadIdx.x;
  const int FQ = F / 4;
  int n = idx / FQ;
  if (n >= N_NODES) return;
  int f = (idx - n * FQ) * 4;
  float w = dinv[n];
  w *= w;
  const float4 v = *(const float4*)(t + (size_t)n * F + f);
  float4 o;
  o.x = v.x * w + b[f + 0];
  o.y = v.y * w + b[f + 1];
  o.z = v.z * w + b[f + 2];
  o.w = v.w * w + b[f + 3];
  *(float4*)(agg + (size_t)n * F + f) = o;
}

// ---------------------------------------------------------------------------
// Edge scatter: agg[dst[e],f] += t[src[e],f] * dinv[src]*dinv[dst]
// Lane layout: consecutive lanes -> consecutive float4 of one edge (coalesced).
// ---------------------------------------------------------------------------
template <int F>
__global__ void scatter_add_kernel(const float* __restrict__ t,
                                   const int* __restrict__ src,
                                   const int* __restrict__ dst,
                                   const float* __restrict__ dinv,
                                   float* __restrict__ agg) {
  int idx = blockIdx.x * blockDim.x + threadIdx.x;
  const int FQ = F / 4;
  int e = idx / FQ;
  if (e >= N_EDGES) return;
  int f = (idx - e * FQ) * 4;
  int s = src[e], d = dst[e];
  float w = dinv[s] * dinv[d];
  const float4 v = *(const float4*)(t + (size_t)s * F + f);
  float* p = agg + (size_t)d * F + f;
  atomicAdd(p + 0, v.x * w);
  atomicAdd(p + 1, v.y * w);
  atomicAdd(p + 2, v.z * w);
  atomicAdd(p + 3, v.w * w);
}

// ---------------------------------------------------------------------------
// Edge GEMM0: Y[e,0:64] = concat(h[src[e]], h[dst[e]]) @ W_m0[128,64] + b
// A-fragments gathered straight from h3 (L2-resident). Accumulates per-column
// sum / sum-of-squares for training-mode BatchNorm (LDS reduce -> global atomics).
// 50000 m-tiles = MPW(5) * 8 waves * 1250 blocks exactly -> no partial blocks,
// so __syncthreads for the BN reduction is safe.
// ---------------------------------------------------------------------------
__global__ __launch_bounds__(256) void edge_gemm0_kernel(
    const float* __restrict__ H, const int* __restrict__ src,
    const int* __restrict__ dst, const float* __restrict__ W,
    const float* __restrict__ bias, float* __restrict__ Y,
    float* __restrict__ gsum, float* __restrict__ gsq) {
  __shared__ float lsum[16], lsq[16];
  if (threadIdx.x < 16) { lsum[threadIdx.x] = 0.0f; lsq[threadIdx.x] = 0.0f; }
  __syncthreads();

  const int lane  = threadIdx.x & 31;
  const int mBase = (blockIdx.x * 8 + (threadIdx.x >> 5)) * MPW;
  const int col   = blockIdx.y * 16 + (lane & 15);
  const int kh    = (lane >> 4) * 2;

  const float* __restrict__ ps[MPW];
  const float* __restrict__ pd[MPW];
#pragma unroll
  for (int i = 0; i < MPW; ++i) {
    int row = (mBase + i) * 16 + (lane & 15);      // edge id
    ps[i] = H + (size_t)src[row] * OUTF;
    pd[i] = H + (size_t)dst[row] * OUTF;
  }

  v8f acc[MPW];
#pragma unroll
  for (int i = 0; i < MPW; ++i) acc[i] = (v8f){};

#pragma unroll
  for (int k = 0; k < OUTF; k += 4) {              // first half: h[src]
    v2f b = {W[(k + kh) * OUTF + col], W[(k + kh + 1) * OUTF + col]};
#pragma unroll
    for (int i = 0; i < MPW; ++i) {
      v2f a = {ps[i][k + kh], ps[i][k + kh + 1]};
      acc[i] = wmma_f32(a, b, acc[i]);
    }
  }
#pragma unroll
  for (int k = 0; k < OUTF; k += 4) {              // second half: h[dst]
    v2f b = {W[(OUTF + k + kh) * OUTF + col], W[(OUTF + k + kh + 1) * OUTF + col]};
#pragma unroll
    for (int i = 0; i < MPW; ++i) {
      v2f a = {pd[i][k + kh], pd[i][k + kh + 1]};
      acc[i] = wmma_f32(a, b, acc[i]);
    }
  }

  const float bb = bias[col];
  float s8 = 0.0f, q8 = 0.0f;
#pragma unroll
  for (int i = 0; i < MPW; ++i) {
    const int rbase = (mBase + i) * 16 + (lane >> 4) * 8;
#pragma unroll
    for (int r = 0; r < 8; ++r) {
      float v = acc[i][r] + bb;
      Y[(size_t)(rbase + r) * OUTF + col] = v;
      s8 += v;
      q8 += v * v;
    }
  }
  atomicAdd(&lsum[lane & 15], s8);
  atomicAdd(&lsq[lane & 15], q8);
  __syncthreads();
  if (threadIdx.x < 16) {
    atomicAdd(&gsum[blockIdx.y * 16 + threadIdx.x], lsum[threadIdx.x]);
    atomicAdd(&gsq[blockIdx.y * 16 + threadIdx.x], lsq[threadIdx.x]);
  }
}

// ---------------------------------------------------------------------------
// BN param fold: scale = g * rsqrt(var + eps); shift = beta - mean * scale
// ---------------------------------------------------------------------------
__global__ void bn_params_kernel(const float* __restrict__ sum,
                                 const float* __restrict__ sq,
                                 const float* __restrict__ g,
                                 const float* __restrict__ be,
                                 float* __restrict__ scale,
                                 float* __restrict__ shift, int F, float invE) {
  int i = threadIdx.x;
  if (i < F) {
    float mu  = sum[i] * invE;
    float var = sq[i] * invE - mu * mu;
    float sc  = g[i] * rsqrtf(var + BN_EPS);
    scale[i] = sc;
    shift[i] = be[i] - mu * sc;
  }
}

// ---------------------------------------------------------------------------
// Edge GEMM1: Y1[e,0:32] = relu(bn(Y0[e,:])) @ W_m1[64,32] + b ; BN stats again.
// ---------------------------------------------------------------------------
__global__ __launch_bounds__(256) void edge_gemm1_kernel(
    const float* __restrict__ Y0, const float* __restrict__ scale,
    const float* __restrict__ shift, const float* __restrict__ W,
    const float* __restrict__ bias, float* __restrict__ Y1,
    float* __restrict__ gsum, float* __restrict__ gsq) {
  __shared__ float lsum[16], lsq[16];
  if (threadIdx.x < 16) { lsum[threadIdx.x] = 0.0f; lsq[threadIdx.x] = 0.0f; }
  __syncthreads();

  const int lane  = threadIdx.x & 31;
  const int mBase = (blockIdx.x * 8 + (threadIdx.x >> 5)) * MPW;
  const int col   = blockIdx.y * 16 + (lane & 15);
  const int kh    = (lane >> 4) * 2;

  const float* __restrict__ ap[MPW];
#pragma unroll
  for (int i = 0; i < MPW; ++i)
    ap[i] = Y0 + (size_t)((mBase + i) * 16 + (lane & 15)) * OUTF;

  v8f acc[MPW];
#pragma unroll
  for (int i = 0; i < MPW; ++i) acc[i] = (v8f){};

#pragma unroll
  for (int k = 0; k < OUTF; k += 4) {
    const int j0 = k + kh, j1 = k + kh + 1;
    const float sc0 = scale[j0], sh0 = shift[j0];
    const float sc1 = scale[j1], sh1 = shift[j1];
    v2f b = {W[j0 * 32 + col], W[j1 * 32 + col]};
#pragma unroll
    for (int i = 0; i < MPW; ++i) {
      v2f a = {fmaxf(ap[i][j0] * sc0 + sh0, 0.0f),
               fmaxf(ap[i][j1] * sc1 + sh1, 0.0f)};
      acc[i] = wmma_f32(a, b, acc[i]);
    }
  }

  const float bb = bias[col];
  float s8 = 0.0f, q8 = 0.0f;
#pragma unroll
  for (int i = 0; i < MPW; ++i) {
    const int rbase = (mBase + i) * 16 + (lane >> 4) * 8;
#pragma unroll
    for (int r = 0; r < 8; ++r) {
      float v = acc[i][r] + bb;
      Y1[(size_t)(rbase + r) * 32 + col] = v;
      s8 += v;
      q8 += v * v;
    }
  }
  atomicAdd(&lsum[lane & 15], s8);
  atomicAdd(&lsq[lane & 15], q8);
  __syncthreads();
  if (threadIdx.x < 16) {
    atomicAdd(&gsum[blockIdx.y * 16 + threadIdx.x], lsum[threadIdx.x]);
    atomicAdd(&gsq[blockIdx.y * 16 + threadIdx.x], lsq[threadIdx.x]);
  }
}

// ---------------------------------------------------------------------------
// Final: out[e] = sigmoid( relu(bn(Y1[e,:])) . W_m2 + b_m2 )
// ---------------------------------------------------------------------------
__global__ __launch_bounds__(256) void edge_final_kernel(
    const float* __restrict__ Y1, const float* __restrict__ scale,
    const float* __restrict__ shift, const float* __restrict__ W2,
    const float* __restrict__ b2, float* __restrict__ out) {
  int e = blockIdx.x * blockDim.x + threadIdx.x;
  if (e >= N_EDGES) return;
  const float* __restrict__ y = Y1 + (size_t)e * 32;
  float d = b2[0];
#pragma unroll
  for (int j = 0; j < 32; ++j)
    d += fmaxf(y[j] * scale[j] + shift[j], 0.0f) * W2[j];
  out[e] = 1.0f / (1.0f + expf(-d));
}

__global__ void zero_kernel(float* __restrict__ p, int n) {
  int i = blockIdx.x * blockDim.x + threadIdx.x;
  if (i < n) p[i] = 0.0f;
}

// ---------------------------------------------------------------------------
// Host-side orchestration (graph-capture safe: only kernel launches on stream)
// ---------------------------------------------------------------------------
extern "C" void kernel_launch(void* const* d_in, const int* in_sizes, int n_in,
                              void* d_out, int out_size, void* d_ws, size_t ws_size,
                              hipStream_t stream) {
  (void)in_sizes; (void)n_in; (void)out_size; (void)ws_size;
  const float* x     = (const float*)d_in[0];
  const int*   ei    = (const int*)d_in[1];
  const float* W_emb = (const float*)d_in[2];
  const float* b_emb = (const float*)d_in[3];
  const float* W_g1  = (const float*)d_in[4];
  const float* b_g1  = (const float*)d_in[5];
  const float* W_g2  = (const float*)d_in[6];
  const float* b_g2  = (const float*)d_in[7];
  const float* W_m0  = (const float*)d_in[8];
  const float* b_m0  = (const float*)d_in[9];
  const float* g_m0  = (const float*)d_in[10];
  const float* be_m0 = (const float*)d_in[11];
  const float* W_m1  = (const float*)d_in[12];
  const float* b_m1  = (const float*)d_in[13];
  const float* g_m1  = (const float*)d_in[14];
  const float* be_m1 = (const float*)d_in[15];
  const float* W_m2  = (const float*)d_in[16];
  const float* b_m2  = (const float*)d_in[17];

  const int* src = ei;             // edge_index[0]
  const int* dst = ei + N_EDGES;   // edge_index[1]
  float* out = (float*)d_out;

  // Workspace layout (floats), with buffer reuse:
  float* ws = (float*)d_ws;
  size_t o = 0;
  float* deg  = ws + o; o += N_NODES;
  float* dinv = ws + o; o += N_NODES;
  float* h1   = ws + o; o += (size_t)N_NODES * HID;   // later reused as t2
  float* t1   = ws + o; o += (size_t)N_NODES * HID;   // later reused as h3 (agg2)
  float* agg1 = ws + o;                                // overlaid by y0 afterwards
  float* y0   = agg1;   o += (size_t)N_EDGES * OUTF;   // [E,64]
  float* y1   = ws + o; o += (size_t)N_EDGES * 32;     // [E,32]
  float* sum0   = ws + o;        // 64
  float* sq0    = sum0 + 64;     // 64
  float* sum1   = sum0 + 128;    // 32
  float* sq1    = sum0 + 160;    // 32
  float* scale0 = sum0 + 192;    // 64
  float* shift0 = sum0 + 256;    // 64
  float* scale1 = sum0 + 320;    // 32
  float* shift1 = sum0 + 352;    // 32
  float* t2   = h1;
  float* h3   = t1;

  const int MT_NODE   = N_NODES / 16;                    // 3125 m-tiles
  const int WAVE_NODE = (MT_NODE + MPW - 1) / MPW;       // 625 wave-units
  const int NB_NODE   = (WAVE_NODE + 7) / 8;             // 79 blocks
  const int NB_EDGE   = (N_EDGES / 16) / (MPW * 8);      // 1250 blocks (exact)

  // 1) degrees & symmetric-normalization factors
  deg_init_kernel<<<(N_NODES + 255) / 256, 256, 0, stream>>>(deg);
  deg_count_kernel<<<(N_EDGES + 255) / 256, 256, 0, stream>>>(dst, deg);
  dinv_kernel<<<(N_NODES + 255) / 256, 256, 0, stream>>>(deg, dinv);

  // 2) h1 = x @ W_emb + b_emb        [N,128]
  dim3 gEmb(NB_NODE, HID / 16);
  node_gemm_kernel<IN_D, HID, false, true>
      <<<gEmb, 256, 0, stream>>>(x, W_emb, b_emb, h1, N_NODES);

  // 3) t1 = h1 @ W_g1                [N,128]
  node_gemm_kernel<HID, HID, false, false>
      <<<gEmb, 256, 0, stream>>>(h1, W_g1, nullptr, t1, N_NODES);

  // 4) agg1 = t1*dinv^2 + b_g1 ; scatter edges
  self_init_kernel<HID>
      <<<((size_t)N_NODES * (HID / 4) + 255) / 256, 256, 0, stream>>>(t1, dinv, b_g1, agg1);
  scatter_add_kernel<HID>
      <<<((size_t)N_EDGES * (HID / 4) + 255) / 256, 256, 0, stream>>>(t1, src, dst, dinv, agg1);

  // 5) t2 = relu(agg1) @ W_g2        [N,64]  (ReLU fused into A load)
  dim3 gG2(NB_NODE, OUTF / 16);
  node_gemm_kernel<HID, OUTF, true, false>
      <<<gG2, 256, 0, stream>>>(agg1, W_g2, nullptr, t2, N_NODES);

  // 6) h3 = t2*dinv^2 + b_g2 ; scatter edges
  self_init_kernel<OUTF>
      <<<((size_t)N_NODES * (OUTF / 4) + 255) / 256, 256, 0, stream>>>(t2, dinv, b_g2, h3);
  scatter_add_kernel<OUTF>
      <<<((size_t)N_EDGES * (OUTF / 4) + 255) / 256, 256, 0, stream>>>(t2, src, dst, dinv, h3);

  // 7) zero BN statistic accumulators (192 floats)
  zero_kernel<<<1, 256, 0, stream>>>(sum0, 192);

  // 8) y0 = concat(h3[src],h3[dst]) @ W_m0 + b_m0, with BN stats
  dim3 gE0(NB_EDGE, OUTF / 16);
  edge_gemm0_kernel<<<gE0, 256, 0, stream>>>(h3, src, dst, W_m0, b_m0, y0, sum0, sq0);
  bn_params_kernel<<<1, 64, 0, stream>>>(sum0, sq0, g_m0, be_m0, scale0, shift0,
                                         64, 1.0f / (float)N_EDGES);

  // 9) y1 = relu(bn(y0)) @ W_m1 + b_m1, with BN stats
  dim3 gE1(NB_EDGE, 2);
  edge_gemm1_kernel<<<gE1, 256, 0, stream>>>(y0, scale0, shift0, W_m1, b_m1, y1, sum1, sq1);
  bn_params_kernel<<<1, 32, 0, stream>>>(sum1, sq1, g_m1, be_m1, scale1, shift1,
                                         32, 1.0f / (float)N_EDGES);

  // 10) out = sigmoid(relu(bn(y1)) . W_m2 + b_m2)
  edge_final_kernel<<<(N_EDGES + 255) / 256, 256, 0, stream>>>(y1, scale1, shift1,
                                                               W_m2, b_m2, out);
}